// SparseArch_15324443312170
// MI455X (gfx1250) — compile-verified
//
#include <hip/hip_runtime.h>

// Reference constants
#define ZCH_SIZE 2000000
#define NUM_IDS  1000000

constexpr int       BLOCKS          = 4096;
constexpr int       THREADS         = 256;            // 8 wave32 per block
constexpr int       WAVES_PER_BLOCK = THREADS / 32;
constexpr long long TOTAL_ROWS      = 2LL * NUM_IDS;  // 2,000,000
constexpr int       NWAVES          = BLOCKS * WAVES_PER_BLOCK;  // 32768
// rows per wave, rounded up, forced even (we process rows in pairs) -> 62
constexpr int       CHUNK = (int)(((TOTAL_ROWS + NWAVES - 1) / NWAVES + 1) & ~1LL);

// ---------------------------------------------------------------------------
// Kernel 1: fused remap + gather + partial sum.
//  * Each wave owns a contiguous chunk of CHUNK rows. Its id slice is
//    contiguous, so we stage it into LDS with the CDNA5 async global->LDS
//    copy (global_load_async_to_lds_b32, ASYNCcnt-tracked, completed with
//    s_wait_asynccnt). Hot-loop id reads are then LDS hits.
//  * Gather: 16 lanes per row, float4 per lane -> one v-load moves 2 rows
//    (512B) per wave per instruction, perfectly coalesced 256B per row.
//  * global_prefetch_b8 lookahead (4 pairs ahead) hides HBM latency on the
//    random gather (working set > 192MB L2).
//  * Deterministic: fixed assignment + fixed-order reductions, no atomics.
// ---------------------------------------------------------------------------
__global__ __launch_bounds__(THREADS) void gather_sum_kernel(
    const float* __restrict__ t0, const float* __restrict__ t1,
    const int*   __restrict__ ids0, const int* __restrict__ ids1,
    float* __restrict__ block_partials)   // [BLOCKS]
{
    __shared__ int shIds[WAVES_PER_BLOCK][CHUNK];

    const int lane = threadIdx.x & 31;
    const int wib  = threadIdx.x >> 5;
    const long long waveId = (long long)blockIdx.x * WAVES_PER_BLOCK + wib;
    const long long base   = waveId * CHUNK;

    int cnt = 0;
    if (base < TOTAL_ROWS) {
        const long long rem = TOTAL_ROWS - base;
        cnt = (rem < CHUNK) ? (int)rem : CHUNK;      // always even (see constants)
    }

    // ---- stage this wave's ids into LDS via async global->LDS (CDNA5) ----
    for (int k = lane; k < cnt; k += 32) {
        const long long row = base + k;
        const int* src = (row < NUM_IDS) ? (ids0 + row) : (ids1 + (row - NUM_IDS));
        const unsigned ldsOff = (unsigned)(uintptr_t)&shIds[wib][k];   // LDS byte addr
        asm volatile("global_load_async_to_lds_b32 %0, %1, off"
                     :: "v"(ldsOff), "v"((unsigned long long)(uintptr_t)src)
                     : "memory");
    }
    asm volatile("s_wait_asynccnt 0x0" ::: "memory");   // own-wave slice only -> no barrier

    // ---- gather + accumulate: 2 rows per wave per iteration ----
    float a0 = 0.f, a1 = 0.f, a2 = 0.f, a3 = 0.f;
    const int sub  = lane >> 4;          // 0/1: which row of the pair
    const int eoff = (lane & 15) * 4;    // float offset within the row

    for (int k = 0; k < cnt; k += 2) {
        // prefetch 4 pairs ahead (ids are in LDS, address calc nearly free)
        const int kp = k + 8;
        if (kp < cnt) {
            const int       rkp  = kp + sub;
            const long long rowp = base + rkp;
            const int       rp   = shIds[wib][rkp] % ZCH_SIZE;
            const float*    tbp  = (rowp < NUM_IDS) ? t0 : t1;
            __builtin_prefetch(tbp + (size_t)rp * 64u + eoff, 0, 3);
        }

        const int       rk  = k + sub;
        const long long row = base + rk;
        const int       r   = shIds[wib][rk] % ZCH_SIZE;   // managed-collision remap
        const float*    tb  = (row < NUM_IDS) ? t0 : t1;
        const float4 v = *reinterpret_cast<const float4*>(tb + (size_t)r * 64u + eoff);
        a0 += v.x; a1 += v.y; a2 += v.z; a3 += v.w;
    }

    // deterministic wave32 tree reduction
    float acc = (a0 + a1) + (a2 + a3);
    #pragma unroll
    for (int off = 16; off > 0; off >>= 1)
        acc += __shfl_down(acc, off, 32);

    __shared__ float wsum[WAVES_PER_BLOCK];
    if (lane == 0) wsum[wib] = acc;
    __syncthreads();

    if (threadIdx.x == 0) {
        float b = 0.f;
        #pragma unroll
        for (int i = 0; i < WAVES_PER_BLOCK; ++i) b += wsum[i];
        block_partials[blockIdx.x] = b;   // fully overwritten every call
    }
}

// ---------------------------------------------------------------------------
// Kernel 2: coalesced remap output. out[1+i] = float(ids[i] % ZCH).
// Non-temporal stores: streaming output, keep L2 for the gather working set.
// ---------------------------------------------------------------------------
__global__ __launch_bounds__(256) void remap_kernel(
    const int* __restrict__ ids0, const int* __restrict__ ids1,
    float* __restrict__ out)
{
    const int i = blockIdx.x * 256 + threadIdx.x;
    if (i < (int)TOTAL_ROWS) {
        const int id = (i < NUM_IDS) ? ids0[i] : ids1[i - NUM_IDS];
        __builtin_nontemporal_store((float)(id % ZCH_SIZE), &out[1 + i]);
    }
}

// ---------------------------------------------------------------------------
// Kernel 3: deterministic fixed-order reduction of block partials -> mean.
// ---------------------------------------------------------------------------
__global__ __launch_bounds__(256) void finalize_kernel(
    const float* __restrict__ block_partials, int nblocks,
    float* __restrict__ out)
{
    __shared__ double sh[256];
    double s = 0.0;
    for (int i = threadIdx.x; i < nblocks; i += 256)
        s += (double)block_partials[i];
    sh[threadIdx.x] = s;
    __syncthreads();
    #pragma unroll
    for (int off = 128; off > 0; off >>= 1) {
        if (threadIdx.x < off) sh[threadIdx.x] += sh[threadIdx.x + off];
        __syncthreads();
    }
    if (threadIdx.x == 0) {
        const double denom = 2.0 * (double)NUM_IDS * 64.0;   // mean over [2N, 64]
        out[0] = (float)(sh[0] / denom);
    }
}

// ---------------------------------------------------------------------------
// Launch wrapper
// ---------------------------------------------------------------------------
extern "C" void kernel_launch(void* const* d_in, const int* in_sizes, int n_in,
                              void* d_out, int out_size, void* d_ws, size_t ws_size,
                              hipStream_t stream) {
    const float* t0   = (const float*)d_in[0];  // table_0 [2M, 64] f32
    const float* t1   = (const float*)d_in[1];  // table_1 [2M, 64] f32
    const int*   ids0 = (const int*)d_in[2];    // ids_0 [1M]
    const int*   ids1 = (const int*)d_in[3];    // ids_1 [1M]
    float* out        = (float*)d_out;          // [1 + 2M] : loss, r0, r1
    float* partials   = (float*)d_ws;           // BLOCKS floats of scratch

    gather_sum_kernel<<<BLOCKS, THREADS, 0, stream>>>(t0, t1, ids0, ids1, partials);

    const int remapBlocks = (int)((TOTAL_ROWS + 255) / 256);
    remap_kernel<<<remapBlocks, 256, 0, stream>>>(ids0, ids1, out);

    finalize_kernel<<<1, 256, 0, stream>>>(partials, BLOCKS, out);
}